// DotGATLayer_3968549782096
// MI455X (gfx1250) — compile-verified
//
#include <hip/hip_runtime.h>
#include <math.h>

#define N_NODES 50000
#define N_EDGES 800000
#define IN_F    128
#define OUT_F   64

typedef __attribute__((ext_vector_type(2))) float v2f;
typedef __attribute__((ext_vector_type(8))) float v8f;

// Order-preserving float <-> uint mapping for atomicMax-based segment max.
__device__ __forceinline__ unsigned fkey(float f) {
    unsigned u = __float_as_uint(f);
    return (u & 0x80000000u) ? ~u : (u | 0x80000000u);
}
__device__ __forceinline__ float funkey(unsigned u) {
    return __uint_as_float((u & 0x80000000u) ? (u & 0x7FFFFFFFu) : ~u);
}

// ---------------------------------------------------------------------------
// Init: out = 0, denom = 0, segment max = key(-inf)
// ---------------------------------------------------------------------------
__global__ void gat_init(unsigned* __restrict__ segbits,
                         float* __restrict__ denom,
                         float* __restrict__ out) {
    int i = blockIdx.x * blockDim.x + threadIdx.x;
    if (i < N_NODES) {
        segbits[i] = fkey(-INFINITY);
        denom[i]   = 0.0f;
    }
    if (i < N_NODES * OUT_F) out[i] = 0.0f;
}

// ---------------------------------------------------------------------------
// Q = x @ Wq^T, K = x @ Wk^T via V_WMMA_F32_16X16X4_F32 (exact f32 path).
// Block = 256 threads = 8 waves; wave w: matrix (w>>2) in {Q,K}, col tile w&3.
// Each block covers 16 rows (nodes). 50000 is a multiple of 16 -> no guards,
// EXEC stays all-ones as WMMA requires.
// ---------------------------------------------------------------------------
__global__ void qk_gemm(const float* __restrict__ x,
                        const float* __restrict__ Wq,
                        const float* __restrict__ Wk,
                        float* __restrict__ Q,
                        float* __restrict__ K) {
    const int lane    = threadIdx.x & 31;
    const int wave    = threadIdx.x >> 5;     // 0..7
    const int mat     = wave >> 2;            // 0 = Q, 1 = K
    const int colBase = (wave & 3) * 16;
    const int rowBase = blockIdx.x * 16;

    const float* __restrict__ W   = mat ? Wk : Wq;
    float*       __restrict__ Out = mat ? K  : Q;

    const int l    = lane & 15;        // M index (A) / N index (B)
    const int koff = (lane >> 4) * 2;  // half-wave selects K pair {0,1} vs {2,3}

    // A: x[rowBase+l, k+koff .. +1] ; B[k][n] = W[n][k] (W is [out,in])
    const float* __restrict__ arow = x + (size_t)(rowBase + l) * IN_F + koff;
    const float* __restrict__ brow = W + (size_t)(colBase + l) * IN_F + koff;

    v8f acc = {};
#pragma unroll 8
    for (int k = 0; k < IN_F; k += 4) {
        v2f a = *(const v2f*)(arow + k);
        v2f b = *(const v2f*)(brow + k);
        acc = __builtin_amdgcn_wmma_f32_16x16x4_f32(
            /*neg_a=*/false, a, /*neg_b=*/false, b,
            /*c_mod=*/(short)0, acc, /*reuse_a=*/false, /*reuse_b=*/false);
    }

    // C/D layout: VGPR r -> M = r (lanes 0-15) or M = 8 + r (lanes 16-31), N = lane&15
    const int outRow = rowBase + ((lane >> 4) ? 8 : 0);
    const int outCol = colBase + l;
#pragma unroll
    for (int r = 0; r < 8; ++r)
        Out[(size_t)(outRow + r) * OUT_F + outCol] = acc[r];
}

// ---------------------------------------------------------------------------
// Per-edge logits: alpha = dot(Q[row], K[col]) / 8 ; segment max via atomicMax
// One wave per edge, 2 features per lane, shfl_xor tree reduction.
// ---------------------------------------------------------------------------
__global__ void edge_logits(const int* __restrict__ ei,
                            const float* __restrict__ Q,
                            const float* __restrict__ K,
                            float* __restrict__ alpha,
                            unsigned* __restrict__ segbits) {
    const int lane = threadIdx.x & 31;
    const int wid  = (blockIdx.x * blockDim.x + threadIdx.x) >> 5;
    const int nw   = (gridDim.x * blockDim.x) >> 5;

    for (int e = wid; e < N_EDGES; e += nw) {
        const int row = ei[e];              // edge_index[0][e]
        const int col = ei[N_EDGES + e];    // edge_index[1][e]
        const float* __restrict__ q = Q + (size_t)row * OUT_F;
        const float* __restrict__ k = K + (size_t)col * OUT_F;
        float s = q[lane] * k[lane] + q[lane + 32] * k[lane + 32];
#pragma unroll
        for (int off = 16; off; off >>= 1) s += __shfl_xor(s, off, 32);
        s *= 0.125f;  // 1/sqrt(OUT_F)
        if (lane == 0) {
            alpha[e] = s;
            atomicMax(segbits + col, fkey(s));
        }
    }
}

// ---------------------------------------------------------------------------
// ex = exp(alpha - segmax[col]); denom[col] += ex  (in-place on alpha buffer)
// ---------------------------------------------------------------------------
__global__ void edge_exp(const int* __restrict__ ei,
                         const unsigned* __restrict__ segbits,
                         float* __restrict__ alpha,
                         float* __restrict__ denom) {
    const int e = blockIdx.x * blockDim.x + threadIdx.x;
    if (e >= N_EDGES) return;
    const int col = ei[N_EDGES + e];
    const float v = expf(alpha[e] - funkey(segbits[col]));
    alpha[e] = v;
    atomicAdd(denom + col, v);
}

// ---------------------------------------------------------------------------
// out[col] += Q[row] * (ex / (denom[col] + 1e-16)) ; wave per edge.
// ---------------------------------------------------------------------------
__global__ void edge_scatter(const int* __restrict__ ei,
                             const float* __restrict__ Q,
                             const float* __restrict__ ex,
                             const float* __restrict__ denom,
                             float* __restrict__ out) {
    const int lane = threadIdx.x & 31;
    const int wid  = (blockIdx.x * blockDim.x + threadIdx.x) >> 5;
    const int nw   = (gridDim.x * blockDim.x) >> 5;

    for (int e = wid; e < N_EDGES; e += nw) {
        const int row = ei[e];
        const int col = ei[N_EDGES + e];
        const float w = ex[e] / (denom[col] + 1e-16f);
        const float* __restrict__ q = Q + (size_t)row * OUT_F;
        float* __restrict__ o = out + (size_t)col * OUT_F;
        atomicAdd(o + lane,      q[lane]      * w);
        atomicAdd(o + lane + 32, q[lane + 32] * w);
    }
}

// ---------------------------------------------------------------------------
extern "C" void kernel_launch(void* const* d_in, const int* in_sizes, int n_in,
                              void* d_out, int out_size, void* d_ws, size_t ws_size,
                              hipStream_t stream) {
    const float* x  = (const float*)d_in[0];
    const int*   ei = (const int*)d_in[1];   // [2, E] row-major: row ids then col ids
    const float* Wq = (const float*)d_in[2];
    const float* Wk = (const float*)d_in[3];
    float* out = (float*)d_out;

    char* ws = (char*)d_ws;
    float*    Q       = (float*)ws;    ws += (size_t)N_NODES * OUT_F * sizeof(float);
    float*    K       = (float*)ws;    ws += (size_t)N_NODES * OUT_F * sizeof(float);
    float*    alpha   = (float*)ws;    ws += (size_t)N_EDGES * sizeof(float);
    unsigned* segbits = (unsigned*)ws; ws += (size_t)N_NODES * sizeof(unsigned);
    float*    denom   = (float*)ws;

    gat_init<<<(N_NODES * OUT_F + 255) / 256, 256, 0, stream>>>(segbits, denom, out);
    qk_gemm<<<N_NODES / 16, 256, 0, stream>>>(x, Wq, Wk, Q, K);
    edge_logits<<<4096, 256, 0, stream>>>(ei, Q, K, alpha, segbits);
    edge_exp<<<(N_EDGES + 255) / 256, 256, 0, stream>>>(ei, segbits, alpha, denom);
    edge_scatter<<<4096, 256, 0, stream>>>(ei, Q, alpha, denom, out);
}